// repulsive_15796889714854
// MI455X (gfx1250) — compile-verified
//
#include <hip/hip_runtime.h>
#include <hip/hip_bf16.h>

typedef float __attribute__((ext_vector_type(4))) v4f;
typedef int   __attribute__((ext_vector_type(4))) v4i;

// Non-returning device-scope FP32 atomic add, explicit CDNA5 encoding:
//   global_atomic_add_f32 vaddr_off, vdata, s[base:base+1] scope:SCOPE_DEV
// vaddr is a 32-bit byte offset, saddr is the 64-bit base (GVS mode).
// Non-returning => tracked on STOREcnt only; no return-data dependency.
__device__ __forceinline__ void atomic_add_f32_dev(float* base, unsigned byte_off, float val) {
    asm volatile("global_atomic_add_f32 %0, %1, %2 scope:SCOPE_DEV"
                 :
                 : "v"(byte_off), "v"(val), "s"(base)
                 : "memory");
}

// ---------------------------------------------------------------------------
// Kernel 1: zero the per-atom accumulator (harness poisons d_out with 0xAA).
// ---------------------------------------------------------------------------
__global__ void repulsive_zero_kernel(float* __restrict__ out, int n) {
    int i = (blockIdx.x * blockDim.x + threadIdx.x) * 4;
    if (i + 3 < n) {
        v4f z = {0.f, 0.f, 0.f, 0.f};
        *reinterpret_cast<v4f*>(out + i) = z;
    } else if (i < n) {
        out[i] = 0.f;
        if (i + 1 < n) out[i + 1] = 0.f;
        if (i + 2 < n) out[i + 2] = 0.f;
    }
}

// ---------------------------------------------------------------------------
// Kernel 2: per-edge energy + scatter-add.
//   contribution(e) = 0.5f * (expf(-dist[e]) - e0),  e0 = expf(-RC)
//   out[ind_2[e][0]] += contribution(e)
// 4 edges per thread:
//   - 1x b128 NT load of dist            (read-once stream)
//   - 2x b128 NT load of ind_2 pairs     (receiving idx at .x/.z of each)
//   - 4x global_atomic_add_f32 scope:DEV (inline asm, non-returning)
// NT hints keep the ~400 MB streaming reads from evicting the 4 MB
// accumulator, which stays resident in the 192 MB L2.
// ---------------------------------------------------------------------------
__global__ void repulsive_edge_kernel(const float* __restrict__ dist,
                                      const int*   __restrict__ ind2,
                                      float*       __restrict__ out,
                                      int n_edges, float e0) {
    int base = (blockIdx.x * blockDim.x + threadIdx.x) * 4;
    if (base + 3 < n_edges) {
        v4f d  = __builtin_nontemporal_load(reinterpret_cast<const v4f*>(dist + base));
        v4i p0 = __builtin_nontemporal_load(reinterpret_cast<const v4i*>(ind2 + 2 * base));
        v4i p1 = __builtin_nontemporal_load(reinterpret_cast<const v4i*>(ind2 + 2 * base + 4));

        float c0 = 0.5f * (expf(-d.x) - e0);
        float c1 = 0.5f * (expf(-d.y) - e0);
        float c2 = 0.5f * (expf(-d.z) - e0);
        float c3 = 0.5f * (expf(-d.w) - e0);

        atomic_add_f32_dev(out, ((unsigned)p0.x) << 2, c0);
        atomic_add_f32_dev(out, ((unsigned)p0.z) << 2, c1);
        atomic_add_f32_dev(out, ((unsigned)p1.x) << 2, c2);
        atomic_add_f32_dev(out, ((unsigned)p1.z) << 2, c3);
    } else {
        for (int e = base; e < n_edges; ++e) {
            float c = 0.5f * (expf(-dist[e]) - e0);
            atomic_add_f32_dev(out, ((unsigned)ind2[2 * e]) << 2, c);
        }
    }
}

extern "C" void kernel_launch(void* const* d_in, const int* in_sizes, int n_in,
                              void* d_out, int out_size, void* d_ws, size_t ws_size,
                              hipStream_t stream) {
    const float* dist = (const float*)d_in[0];     // [n_edges] f32
    // d_in[1] = ind_1 [natom,1] int32 (not needed by the math)
    const int*   ind2 = (const int*)d_in[2];       // [n_edges,2] int32
    float*       out  = (float*)d_out;             // [natom] f32

    const int n_edges = in_sizes[0];
    const int natom   = out_size;

    const float RC = 3.0f, B = 1.0f, A = 1.0f;
    const float e0 = A * expf(-RC / B);

    const int threads = 256;

    // Pass 1: zero the accumulator.
    {
        int elems_per_block = threads * 4;
        int blocks = (natom + elems_per_block - 1) / elems_per_block;
        repulsive_zero_kernel<<<blocks, threads, 0, stream>>>(out, natom);
    }

    // Pass 2: edge scatter-add (stream-ordered after pass 1).
    {
        int edges_per_block = threads * 4;
        int blocks = (n_edges + edges_per_block - 1) / edges_per_block;
        repulsive_edge_kernel<<<blocks, threads, 0, stream>>>(dist, ind2, out,
                                                              n_edges, e0);
    }
}